// STFRM_17179869671
// MI455X (gfx1250) — compile-verified
//
#include <hip/hip_runtime.h>
#include <hip/hip_bf16.h>
#include <math.h>

// ---------------------------------------------------------------------------
// Types for CDNA5 WMMA f32 16x16x4
// ---------------------------------------------------------------------------
typedef float v2f __attribute__((ext_vector_type(2)));
typedef float v8f __attribute__((ext_vector_type(8)));

#define PLANE 65536           // 256*256
#define QPLANE 16384          // 128*128
#define NC 256                // 4 batches * 64 channels
#define FQ  4194304           // NC * QPLANE
#define FF  16777216          // NC * PLANE

// ---------------------------------------------------------------------------
// DCT matrix constants: C[u*S + j] = cos(pi*(j+0.5)*u/S) * s(u)
// ---------------------------------------------------------------------------
__global__ void k_dct_const(float* __restrict__ C, int S) {
    int i = blockIdx.x * 256 + threadIdx.x;
    if (i >= S * S) return;
    int u = i / S, j = i - u * S;
    float scale = (u == 0) ? sqrtf(1.0f / (float)S) : sqrtf(2.0f / (float)S);
    C[i] = cosf(3.14159265358979323846f * ((float)j + 0.5f) * (float)u / (float)S) * scale;
}

// ---------------------------------------------------------------------------
// DWT: x (NC,256,256) -> LL,HL,LH,HH (NC,128,128)
// ---------------------------------------------------------------------------
__global__ __launch_bounds__(256) void k_dwt(const float* __restrict__ x,
        float* __restrict__ LL, float* __restrict__ HL,
        float* __restrict__ LH, float* __restrict__ HH) {
    size_t t = (size_t)blockIdx.x * 256 + threadIdx.x;       // < FQ
    size_t p = t >> 14;
    int i = (int)((t >> 7) & 127), j = (int)(t & 127);
    const float* xp = x + p * (size_t)PLANE;
    float x1 = xp[(2 * i) * 256 + 2 * j];
    float x2 = xp[(2 * i + 1) * 256 + 2 * j];
    float x3 = xp[(2 * i) * 256 + 2 * j + 1];
    float x4 = xp[(2 * i + 1) * 256 + 2 * j + 1];
    LL[t] = (x1 + x2 + x3 + x4) * 0.25f;
    HL[t] = (-x1 - x2 + x3 + x4) * 0.25f;
    LH[t] = (-x1 + x2 - x3 + x4) * 0.25f;
    HH[t] = (x1 - x2 - x3 + x4) * 0.25f;
}

// ---------------------------------------------------------------------------
// Batched WMMA GEMM (fp32 matrix pipe), 32x32 tile per wave (2x2 register
// blocking -> 4 fragment loads feed 4 WMMAs).
//   mode 0: OUT[b] = M (SxS)      @ X[b] (SxS)
//   mode 1: OUT[b] = X[b] (SxS)   @ M^T
// epilogue: 0 = none
//           1 = freq mask:      (row<maskR && col<maskR) -> 0
//           2 = tanh-threshold then freq mask (spatial branch)
// Fragment layouts (f32 16x16x4):
//  A (16x4): lane L<16: A[L][k0..k0+1]; lane L>=16: A[L-16][k0+2..k0+3]
//  B (4x16): lane L<16: B[k0..k0+1][L]; lane L>=16: B[k0+2..k0+3][L-16]
//  C/D     : VGPR r: lanes0-15 -> (m0+r, n0+lane); lanes16-31 -> (m0+r+8, ...)
// ---------------------------------------------------------------------------
__global__ __launch_bounds__(128) void k_gemm_wmma(
        const float* __restrict__ Mmat, const float* __restrict__ X,
        float* __restrict__ OUT, int S, int mode, int epi, int maskR,
        const float* __restrict__ thrp) {
    const int lane = threadIdx.x;               // 0..31
    const int wave = threadIdx.y;               // 0..3
    const int n0 = blockIdx.x * 32;
    const int m0 = (blockIdx.y * 4 + wave) * 32;
    const size_t b = blockIdx.z;
    const float* __restrict__ Xb = X + b * (size_t)S * S;
    float* __restrict__ Ob = OUT + b * (size_t)S * S;

    const int l16 = lane & 15;
    const int lg = lane >> 4;                   // 0 or 1

    v8f acc[2][2];
    for (int i = 0; i < 2; ++i)
        for (int j = 0; j < 2; ++j)
            acc[i][j] = (v8f){0.f, 0.f, 0.f, 0.f, 0.f, 0.f, 0.f, 0.f};

    if (mode == 0) {
        const int arow0 = m0 + l16, arow1 = m0 + 16 + l16;
        const int bcol0 = n0 + l16, bcol1 = n0 + 16 + l16;
        for (int k0 = 0; k0 < S; k0 += 4) {
            int kk = k0 + 2 * lg;
            v2f a0 = *(const v2f*)(Mmat + (size_t)arow0 * S + kk);
            v2f a1 = *(const v2f*)(Mmat + (size_t)arow1 * S + kk);
            v2f b0, b1;
            b0.x = Xb[(size_t)kk * S + bcol0];
            b0.y = Xb[(size_t)(kk + 1) * S + bcol0];
            b1.x = Xb[(size_t)kk * S + bcol1];
            b1.y = Xb[(size_t)(kk + 1) * S + bcol1];
            acc[0][0] = __builtin_amdgcn_wmma_f32_16x16x4_f32(false, a0, false, b0, (short)0, acc[0][0], false, false);
            acc[0][1] = __builtin_amdgcn_wmma_f32_16x16x4_f32(false, a0, false, b1, (short)0, acc[0][1], false, false);
            acc[1][0] = __builtin_amdgcn_wmma_f32_16x16x4_f32(false, a1, false, b0, (short)0, acc[1][0], false, false);
            acc[1][1] = __builtin_amdgcn_wmma_f32_16x16x4_f32(false, a1, false, b1, (short)0, acc[1][1], false, false);
        }
    } else {
        const int arow0 = m0 + l16, arow1 = m0 + 16 + l16;
        const int brow0 = n0 + l16, brow1 = n0 + 16 + l16;
        for (int k0 = 0; k0 < S; k0 += 4) {
            int kk = k0 + 2 * lg;
            v2f a0 = *(const v2f*)(Xb + (size_t)arow0 * S + kk);
            v2f a1 = *(const v2f*)(Xb + (size_t)arow1 * S + kk);
            v2f b0 = *(const v2f*)(Mmat + (size_t)brow0 * S + kk);
            v2f b1 = *(const v2f*)(Mmat + (size_t)brow1 * S + kk);
            acc[0][0] = __builtin_amdgcn_wmma_f32_16x16x4_f32(false, a0, false, b0, (short)0, acc[0][0], false, false);
            acc[0][1] = __builtin_amdgcn_wmma_f32_16x16x4_f32(false, a0, false, b1, (short)0, acc[0][1], false, false);
            acc[1][0] = __builtin_amdgcn_wmma_f32_16x16x4_f32(false, a1, false, b0, (short)0, acc[1][0], false, false);
            acc[1][1] = __builtin_amdgcn_wmma_f32_16x16x4_f32(false, a1, false, b1, (short)0, acc[1][1], false, false);
        }
    }

    float thr = (epi == 2) ? *thrp : 0.0f;
    for (int tm = 0; tm < 2; ++tm) {
        for (int tn = 0; tn < 2; ++tn) {
            const int col = n0 + tn * 16 + l16;
            for (int r = 0; r < 8; ++r) {
                int row = m0 + tm * 16 + r + 8 * lg;
                float v = acc[tm][tn][r];
                if (epi == 2) { v = (fabsf(v) > thr) ? tanhf(v) : v; }
                if (epi >= 1) { if (row < maskR && col < maskR) v = 0.0f; }
                Ob[(size_t)row * S + col] = v;
            }
        }
    }
}

// ---------------------------------------------------------------------------
// IDWT combine: out = idwt(LL*d, HL*d, LH*d, HH*d)
// ---------------------------------------------------------------------------
__global__ __launch_bounds__(256) void k_idwt(const float* __restrict__ LL,
        const float* __restrict__ HL, const float* __restrict__ LH,
        const float* __restrict__ HH, const float* __restrict__ d,
        float* __restrict__ out) {
    size_t t = (size_t)blockIdx.x * 256 + threadIdx.x;       // < FQ
    size_t p = t >> 14;
    int i = (int)((t >> 7) & 127), j = (int)(t & 127);
    float dv = d[t];
    float a = LL[t] * dv, b2 = HL[t] * dv, c2 = LH[t] * dv, e = HH[t] * dv;
    float x1 = a - b2 - c2 + e;
    float x2 = a - b2 + c2 - e;
    float x3 = a + b2 - c2 - e;
    float x4 = a + b2 + c2 + e;
    float* op = out + p * (size_t)PLANE;
    op[(2 * i) * 256 + 2 * j] = x1;
    op[(2 * i + 1) * 256 + 2 * j] = x2;
    op[(2 * i) * 256 + 2 * j + 1] = x3;
    op[(2 * i + 1) * 256 + 2 * j + 1] = x4;
}

// ---------------------------------------------------------------------------
// Channel gate: per (n,c) plane of dct_full -> 8x8 tile max/mean -> gelu ->
// sum -> relu/sigmoid gate scalar.  One block per plane (256 planes).
// ---------------------------------------------------------------------------
__global__ __launch_bounds__(256) void k_channel_gate(const float* __restrict__ dctf,
        const float* __restrict__ wc1, const float* __restrict__ wc2,
        float* __restrict__ gate) {
    __shared__ float smax[256], ssum[256];
    __shared__ float tgm[64], tga[64];
    int p = blockIdx.x;
    int t = threadIdx.x;
    int tile = t >> 2;          // 0..63
    int sub = t & 3;
    int ti = tile >> 3, tj = tile & 7;
    const float* d = dctf + (size_t)p * PLANE;
    float mx = -3.4e38f, sm = 0.f;
    for (int r = 0; r < 8; ++r) {
        int row = ti * 32 + sub * 8 + r;
        const float* rp = d + row * 256 + tj * 32;
        for (int c = 0; c < 32; ++c) { float v = rp[c]; mx = fmaxf(mx, v); sm += v; }
    }
    smax[t] = mx; ssum[t] = sm;
    __syncthreads();
    if (sub == 0) {
        float m2 = fmaxf(fmaxf(smax[t], smax[t + 1]), fmaxf(smax[t + 2], smax[t + 3]));
        float s2 = ssum[t] + ssum[t + 1] + ssum[t + 2] + ssum[t + 3];
        float avg = s2 * (1.0f / 1024.0f);
        const float k = 0.70710678118654752f;
        tgm[tile] = 0.5f * m2 * (1.0f + erff(m2 * k));
        tga[tile] = 0.5f * avg * (1.0f + erff(avg * k));
    }
    __syncthreads();
    for (int s = 32; s >= 1; s >>= 1) {
        if (t < s) { tgm[t] += tgm[t + s]; tga[t] += tga[t + s]; }
        __syncthreads();
    }
    if (t == 0) {
        int c = p & 63;
        float a = wc1[c] * tgm[0];
        float b = wc1[c] * tga[0];
        float channel = fmaxf(a, 0.f) + fmaxf(b, 0.f);
        float g = fmaxf(wc2[c] * channel, 0.f);
        gate[p] = 1.0f / (1.0f + expf(-g));
    }
}

// ---------------------------------------------------------------------------
// pre_conv = spatial + x * gate[plane]
// ---------------------------------------------------------------------------
__global__ __launch_bounds__(256) void k_addgate(const float* __restrict__ spat,
        const float* __restrict__ x, const float* __restrict__ gate,
        float* __restrict__ out) {
    size_t i = (size_t)blockIdx.x * 256 + threadIdx.x;       // < FF
    size_t p = i >> 16;
    out[i] = spat[i] + x[i] * gate[p];
}

// ---------------------------------------------------------------------------
// 3x3 conv 64->64 as implicit GEMM on the fp32 WMMA pipe.
//   M = 64 output channels, N = 256 pixels of one row h, K = 64*9 = 576.
//   A = w_out viewed as row-major (64 x 576) in k = i*9+dy*3+dx order.
//   B[k][w] = in[n, i, h+dy-1, w+dx-1], zero at SAME-padding borders.
// Bias + ReLU fused in the epilogue.  One wave = one 16x16 tile;
// 4 waves/block cover all 4 output-channel tiles of one (n,h,wtile).
// ---------------------------------------------------------------------------
__global__ __launch_bounds__(128) void k_conv_wmma(const float* __restrict__ in,
        const float* __restrict__ w, const float* __restrict__ bias,
        float* __restrict__ out) {
    const int lane = threadIdx.x;
    const int wave = threadIdx.y;
    const int l16 = lane & 15;
    const int lg = lane >> 4;
    const int n0 = blockIdx.x * 16;     // pixel (w) tile
    const int h = blockIdx.y;
    const int n = blockIdx.z;
    const int m0 = wave * 16;           // output channel tile
    const float* __restrict__ inb = in + (size_t)n * 64 * PLANE;

    v8f acc = {0.f, 0.f, 0.f, 0.f, 0.f, 0.f, 0.f, 0.f};

    const int arow = m0 + l16;
    const int bcol = n0 + l16;
    for (int k0 = 0; k0 < 576; k0 += 4) {
        int kk = k0 + 2 * lg;
        v2f a = *(const v2f*)(w + (size_t)arow * 576 + kk);
        v2f bb;
        {
            int k = kk;
            int i = k / 9, r = k - i * 9;
            int dy = r / 3, dx = r - dy * 3;
            int hy = h + dy - 1, wx = bcol + dx - 1;
            float v = 0.f;
            if ((unsigned)hy < 256u && (unsigned)wx < 256u)
                v = inb[(size_t)i * PLANE + hy * 256 + wx];
            bb.x = v;
        }
        {
            int k = kk + 1;
            int i = k / 9, r = k - i * 9;
            int dy = r / 3, dx = r - dy * 3;
            int hy = h + dy - 1, wx = bcol + dx - 1;
            float v = 0.f;
            if ((unsigned)hy < 256u && (unsigned)wx < 256u)
                v = inb[(size_t)i * PLANE + hy * 256 + wx];
            bb.y = v;
        }
        acc = __builtin_amdgcn_wmma_f32_16x16x4_f32(
            false, a, false, bb, (short)0, acc, false, false);
    }

    for (int r = 0; r < 8; ++r) {
        int o = m0 + r + 8 * lg;
        float v = fmaxf(acc[r] + bias[o], 0.f);
        out[((size_t)(n * 64 + o) * 256 + h) * 256 + (n0 + l16)] = v;
    }
}

// ---------------------------------------------------------------------------
// Per-plane channel_norm in place; gamma/beta index = plane % gmod
// ---------------------------------------------------------------------------
__global__ __launch_bounds__(256) void k_plane_norm(float* __restrict__ y,
        const float* __restrict__ gw, const float* __restrict__ gb, int gmod) {
    __shared__ float s1[256], s2[256];
    int p = blockIdx.x, t = threadIdx.x;
    float* yp = y + (size_t)p * PLANE;
    float a = 0.f, b = 0.f;
    for (int i = t; i < PLANE; i += 256) { float v = yp[i]; a += v; b += v * v; }
    s1[t] = a; s2[t] = b;
    __syncthreads();
    for (int s = 128; s; s >>= 1) {
        if (t < s) { s1[t] += s1[t + s]; s2[t] += s2[t + s]; }
        __syncthreads();
    }
    float mu = s1[0] * (1.0f / PLANE);
    float var = s2[0] * (1.0f / PLANE) - mu * mu;
    float rs = rsqrtf(var + 1e-5f);
    int c = p % gmod;
    float g = gw[c], bb = gb[c];
    for (int i = t; i < PLANE; i += 256) yp[i] = (yp[i] - mu) * rs * g + bb;
}

// ---------------------------------------------------------------------------
// EMA: row / column means per plane
// ---------------------------------------------------------------------------
__global__ __launch_bounds__(256) void k_ema_pool(const float* __restrict__ y,
        float* __restrict__ xh, float* __restrict__ xw) {
    int p = blockIdx.x, t = threadIdx.x;
    const float* yp = y + (size_t)p * PLANE;
    float sr = 0.f, sc = 0.f;
    for (int j = 0; j < 256; ++j) { sr += yp[t * 256 + j]; sc += yp[j * 256 + t]; }
    xh[p * 256 + t] = sr * (1.0f / 256.0f);
    xw[p * 256 + t] = sc * (1.0f / 256.0f);
}

// ---------------------------------------------------------------------------
// EMA: 1x1 conv over cg=8 on pooled vectors, fused sigmoid
// ---------------------------------------------------------------------------
__global__ __launch_bounds__(256) void k_ema_1x1(const float* __restrict__ xh,
        const float* __restrict__ xw, const float* __restrict__ w1,
        const float* __restrict__ b1, float* __restrict__ sh, float* __restrict__ sw) {
    int idx = blockIdx.x * 256 + threadIdx.x;    // (g*8+o)*256 + s, total 65536
    int s = idx & 255;
    int o = (idx >> 8) & 7;
    int g = idx >> 11;
    float ah = b1[o], aw = b1[o];
    for (int i = 0; i < 8; ++i) {
        float wv = w1[o * 8 + i];
        ah += wv * xh[((g * 8 + i) << 8) + s];
        aw += wv * xw[((g * 8 + i) << 8) + s];
    }
    sh[idx] = 1.0f / (1.0f + expf(-ah));
    sw[idx] = 1.0f / (1.0f + expf(-aw));
}

// ---------------------------------------------------------------------------
// EMA: x1 = channel_norm(gx * sh[row] * sw[col]); one block per (g,cc) plane
// ---------------------------------------------------------------------------
__global__ __launch_bounds__(256) void k_ema_x1(const float* __restrict__ gx,
        const float* __restrict__ sh, const float* __restrict__ sw,
        const float* __restrict__ gnw, const float* __restrict__ gnb,
        float* __restrict__ x1) {
    __shared__ float s1[256], s2[256];
    int p = blockIdx.x, t = threadIdx.x, cc = p & 7;
    const float* gp = gx + (size_t)p * PLANE;
    const float* shp = sh + p * 256;
    const float* swp = sw + p * 256;
    float* xp = x1 + (size_t)p * PLANE;
    float a = 0.f, b = 0.f;
    for (int i = t; i < PLANE; i += 256) {
        int r = i >> 8, c = i & 255;
        float v = gp[i] * shp[r] * swp[c];
        xp[i] = v; a += v; b += v * v;
    }
    s1[t] = a; s2[t] = b;
    __syncthreads();
    for (int s = 128; s; s >>= 1) {
        if (t < s) { s1[t] += s1[t + s]; s2[t] += s2[t + s]; }
        __syncthreads();
    }
    float mu = s1[0] * (1.0f / PLANE);
    float var = s2[0] * (1.0f / PLANE) - mu * mu;
    float rs = rsqrtf(var + 1e-5f);
    float g = gnw[cc], bb = gnb[cc];
    for (int i = t; i < PLANE; i += 256) xp[i] = (xp[i] - mu) * rs * g + bb;
}

// ---------------------------------------------------------------------------
// EMA: grouped 3x3 conv (8 ch per group) + bias -> x2  (M=8 < 16: stays VALU)
// ---------------------------------------------------------------------------
__global__ __launch_bounds__(256) void k_ema_conv3(const float* __restrict__ gx,
        const float* __restrict__ w3, const float* __restrict__ b3,
        float* __restrict__ x2) {
    int bid = blockIdx.x;       // (g,o,h) : 32*8*256 = 65536
    int h = bid & 255;
    int go = bid >> 8;
    int o = go & 7, g = go >> 3;
    int wc = threadIdx.x;
    float acc = b3[o];
    for (int i = 0; i < 8; ++i) {
        const float* ip = gx + (size_t)(g * 8 + i) * PLANE;
        const float* wp = w3 + (o * 8 + i) * 9;
        for (int dy = 0; dy < 3; ++dy) {
            int hy = h + dy - 1;
            if (hy < 0 || hy > 255) continue;
            const float* rp = ip + hy * 256;
            float l = (wc > 0) ? rp[wc - 1] : 0.f;
            float m = rp[wc];
            float r = (wc < 255) ? rp[wc + 1] : 0.f;
            acc += wp[dy * 3] * l + wp[dy * 3 + 1] * m + wp[dy * 3 + 2] * r;
        }
    }
    x2[((size_t)go * 256 + h) * 256 + wc] = acc;
}

// ---------------------------------------------------------------------------
// Per-plane mean
// ---------------------------------------------------------------------------
__global__ __launch_bounds__(256) void k_plane_mean(const float* __restrict__ src,
        float* __restrict__ means) {
    __shared__ float s1[256];
    int p = blockIdx.x, t = threadIdx.x;
    const float* sp = src + (size_t)p * PLANE;
    float a = 0.f;
    for (int i = t; i < PLANE; i += 256) a += sp[i];
    s1[t] = a;
    __syncthreads();
    for (int s = 128; s; s >>= 1) {
        if (t < s) s1[t] += s1[t + s];
        __syncthreads();
    }
    if (t == 0) means[p] = s1[0] * (1.0f / PLANE);
}

// ---------------------------------------------------------------------------
// Softmax over 8 channels per group (32 groups)
// ---------------------------------------------------------------------------
__global__ void k_softmax8(const float* __restrict__ m1, const float* __restrict__ m2,
        float* __restrict__ x11, float* __restrict__ x21) {
    int idx = threadIdx.x;      // 256
    int g = idx >> 3;
    float mx1 = -3.4e38f, mx2 = -3.4e38f;
    for (int i = 0; i < 8; ++i) {
        mx1 = fmaxf(mx1, m1[g * 8 + i]);
        mx2 = fmaxf(mx2, m2[g * 8 + i]);
    }
    float s1 = 0.f, s2 = 0.f;
    for (int i = 0; i < 8; ++i) {
        s1 += expf(m1[g * 8 + i] - mx1);
        s2 += expf(m2[g * 8 + i] - mx2);
    }
    x11[idx] = expf(m1[idx] - mx1) / s1;
    x21[idx] = expf(m2[idx] - mx2) / s2;
}

// ---------------------------------------------------------------------------
// Final: weights = x11·x2 + x21·x1 ; out = gx * sigmoid(weights)
// ---------------------------------------------------------------------------
__global__ __launch_bounds__(256) void k_ema_final(const float* __restrict__ gx,
        const float* __restrict__ x1b, const float* __restrict__ x2b,
        const float* __restrict__ x11, const float* __restrict__ x21,
        float* __restrict__ out) {
    size_t idx = (size_t)blockIdx.x * 256 + threadIdx.x;     // g*PLANE + pix
    int g = (int)(idx >> 16);
    int pix = (int)(idx & 65535);
    float wsum = 0.f;
    for (int c = 0; c < 8; ++c) {
        size_t pp = (size_t)(g * 8 + c) * PLANE + pix;
        wsum += x11[g * 8 + c] * x2b[pp] + x21[g * 8 + c] * x1b[pp];
    }
    float s = 1.0f / (1.0f + expf(-wsum));
    for (int cc = 0; cc < 8; ++cc) {
        size_t pp = (size_t)(g * 8 + cc) * PLANE + pix;
        out[pp] = gx[pp] * s;
    }
}

// ---------------------------------------------------------------------------
// Launch
// ---------------------------------------------------------------------------
extern "C" void kernel_launch(void* const* d_in, const int* in_sizes, int n_in,
                              void* d_out, int out_size, void* d_ws, size_t ws_size,
                              hipStream_t stream) {
    const float* x    = (const float*)d_in[0];
    const float* thr  = (const float*)d_in[1];
    const float* wc1  = (const float*)d_in[2];
    const float* wc2  = (const float*)d_in[3];
    const float* wout = (const float*)d_in[4];
    const float* bout = (const float*)d_in[5];
    const float* gnow = (const float*)d_in[6];
    const float* gnob = (const float*)d_in[7];
    const float* ew1  = (const float*)d_in[8];
    const float* eb1  = (const float*)d_in[9];
    const float* ew3  = (const float*)d_in[10];
    const float* eb3  = (const float*)d_in[11];
    const float* egw  = (const float*)d_in[12];
    const float* egb  = (const float*)d_in[13];
    float* out = (float*)d_out;
    float* ws = (float*)d_ws;

    // workspace layout (floats)
    float* C128 = ws;                       // 16384
    float* C256 = C128 + 16384;             // 65536
    float* Qb   = C256 + 65536;             // 6 * FQ quarter buffers; reused as tF
    float* LL = Qb;
    float* HL = Qb + (size_t)FQ;
    float* LH = Qb + (size_t)2 * FQ;
    float* HH = Qb + (size_t)3 * FQ;
    float* tQ = Qb + (size_t)4 * FQ;
    float* dQ = Qb + (size_t)5 * FQ;
    float* tF = Qb;                         // reuse of 6*FQ region (FF < 6*FQ)
    float* dF = Qb + (size_t)6 * FQ;        // FF
    float* sp = dF + (size_t)FF;            // FF (spatial, later x1)
    float* sm = sp + (size_t)FF;            // small region
    float* xh   = sm;                       // 65536
    float* xw   = xh + 65536;               // 65536
    float* sh   = xw + 65536;               // 65536
    float* sw   = sh + 65536;               // 65536
    float* gate = sw + 65536;               // 256
    float* mn1  = gate + 256;               // 256
    float* mn2  = mn1 + 256;                // 256
    float* x11  = mn2 + 256;                // 256
    float* x21  = x11 + 256;                // 256

    dim3 wblk(32, 4);

    // DCT constants
    k_dct_const<<<64, 256, 0, stream>>>(C128, 128);
    k_dct_const<<<256, 256, 0, stream>>>(C256, 256);

    // -------- spatial branch --------
    k_dwt<<<FQ / 256, 256, 0, stream>>>(x, LL, HL, LH, HH);
    // d = dct2(LL) with tanh-threshold + mask fused
    k_gemm_wmma<<<dim3(4, 1, NC), wblk, 0, stream>>>(C128, LL, tQ, 128, 0, 0, 0, thr);
    k_gemm_wmma<<<dim3(4, 1, NC), wblk, 0, stream>>>(C128, tQ, dQ, 128, 1, 2, 32, thr);
    // dct_ = idct2(d)  (same contraction: C · d · C^T)
    k_gemm_wmma<<<dim3(4, 1, NC), wblk, 0, stream>>>(C128, dQ, tQ, 128, 0, 0, 0, thr);
    k_gemm_wmma<<<dim3(4, 1, NC), wblk, 0, stream>>>(C128, tQ, dQ, 128, 1, 0, 0, thr);
    k_idwt<<<FQ / 256, 256, 0, stream>>>(LL, HL, LH, HH, dQ, sp);

    // -------- channel branch (Q buffers dead; reuse as tF) --------
    k_gemm_wmma<<<dim3(8, 2, NC), wblk, 0, stream>>>(C256, x, tF, 256, 0, 0, 0, thr);
    k_gemm_wmma<<<dim3(8, 2, NC), wblk, 0, stream>>>(C256, tF, dF, 256, 1, 1, 64, thr);
    k_gemm_wmma<<<dim3(8, 2, NC), wblk, 0, stream>>>(C256, dF, tF, 256, 0, 0, 0, thr);
    k_gemm_wmma<<<dim3(8, 2, NC), wblk, 0, stream>>>(C256, tF, dF, 256, 1, 0, 0, thr);
    k_channel_gate<<<NC, 256, 0, stream>>>(dF, wc1, wc2, gate);

    // -------- combine + conv (WMMA implicit GEMM) + norm --------
    k_addgate<<<FF / 256, 256, 0, stream>>>(sp, x, gate, tF);
    k_conv_wmma<<<dim3(16, 256, 4), wblk, 0, stream>>>(tF, wout, bout, dF);
    k_plane_norm<<<NC, 256, 0, stream>>>(dF, gnow, gnob, 64);   // dF == y == gx

    // -------- EMA --------
    k_ema_pool<<<NC, 256, 0, stream>>>(dF, xh, xw);
    k_ema_1x1<<<256, 256, 0, stream>>>(xh, xw, ew1, eb1, sh, sw);
    k_ema_x1<<<NC, 256, 0, stream>>>(dF, sh, sw, egw, egb, sp); // x1 -> sp
    k_ema_conv3<<<65536, 256, 0, stream>>>(dF, ew3, eb3, tF);   // x2 -> tF
    k_plane_mean<<<NC, 256, 0, stream>>>(sp, mn1);
    k_plane_mean<<<NC, 256, 0, stream>>>(tF, mn2);
    k_softmax8<<<1, 256, 0, stream>>>(mn1, mn2, x11, x21);
    k_ema_final<<<(32 * PLANE) / 256, 256, 0, stream>>>(dF, sp, tF, x11, x21, out);

    (void)in_sizes; (void)n_in; (void)out_size; (void)ws_size;
}